// MoGRUCopyModel_87351044866253
// MI455X (gfx1250) — compile-verified
//
#include <hip/hip_runtime.h>

// ---- problem constants (from reference) ----
#define B_    32
#define T_    2048
#define H_    256
#define G3    768       // 3*H
#define NV_   18        // embedding rows
#define NO_   17        // head outputs
#define NOP_  32        // padded head cols (2 WMMA N-tiles)

#define HSTR  260       // padded h row stride in LDS (floats) -> bank-conflict free

typedef float v2f __attribute__((ext_vector_type(2)));
typedef float v8f __attribute__((ext_vector_type(8)));

__device__ __forceinline__ float sigf(float x)   { return 1.0f / (1.0f + __expf(-x)); }
__device__ __forceinline__ float tanh_f(float x) { return 2.0f / (1.0f + __expf(-2.0f * x)) - 1.0f; }

// ---------------------------------------------------------------------------
// Prep 1: gi_table[NV][768] = embed @ W_ih + b_ih   (collapses the big GEMM)
// ---------------------------------------------------------------------------
__global__ __launch_bounds__(256)
void prep_gi_table(const float* __restrict__ embed, const float* __restrict__ W_ih,
                   const float* __restrict__ b_ih, float* __restrict__ gi_table)
{
    const int row = blockIdx.x;                 // 0..17
    for (int c = threadIdx.x; c < G3; c += 256) {
        float acc = b_ih[c];
        #pragma unroll 4
        for (int k = 0; k < H_; ++k)
            acc = fmaf(embed[row * H_ + k], W_ih[k * G3 + c], acc);
        gi_table[row * G3 + c] = acc;
    }
}

// ---------------------------------------------------------------------------
// Prep 2: W_hhT[768][256] = transpose(W_hh[256][768])  -> contiguous B-operand
// ---------------------------------------------------------------------------
__global__ __launch_bounds__(256)
void prep_whhT(const float* __restrict__ W_hh, float* __restrict__ W_hhT)
{
    int idx = blockIdx.x * 256 + threadIdx.x;   // n*256 + k
    if (idx < G3 * H_) {
        int n = idx / H_, k = idx - n * H_;
        W_hhT[idx] = W_hh[k * G3 + n];
    }
}

// ---------------------------------------------------------------------------
// Prep 3: W_headT[32][256], rows >= 17 zero-filled
// ---------------------------------------------------------------------------
__global__ __launch_bounds__(256)
void prep_wheadT(const float* __restrict__ W_head, float* __restrict__ W_headT)
{
    int idx = blockIdx.x * 256 + threadIdx.x;   // n*256 + k
    if (idx < NOP_ * H_) {
        int n = idx / H_, k = idx - n * H_;
        W_headT[idx] = (n < NO_) ? W_head[k * NO_ + n] : 0.0f;
    }
}

// ---------------------------------------------------------------------------
// Recurrent scan. grid = 2 workgroups (16 batch rows each), 512 threads = 16 waves.
// Wave w owns h-columns [16w, 16w+16): 3 gate tiles via V_WMMA_F32_16X16X4_F32.
// Waves 0,1 additionally compute the 17-wide head per step.
// ---------------------------------------------------------------------------
__global__ __launch_bounds__(512)
void mogru_scan(const int*   __restrict__ tokens,
                const float* __restrict__ gi_table,
                const float* __restrict__ W_hhT,
                const float* __restrict__ b_hh,
                const float* __restrict__ W_headT,
                const float* __restrict__ b_head,
                float* __restrict__ out_logits,
                float* __restrict__ out_betas)
{
    __shared__ float hbuf[2][16 * HSTR];
    __shared__ int   tokbuf[16];

    const int tid   = threadIdx.x;
    const int wave  = tid >> 5;            // 0..15  (wave32)
    const int lane  = tid & 31;
    const int ln    = lane & 15;           // column-within-tile / M row
    const int khalf = (lane >> 4) * 2;     // K sub-pair per lane half (A/B layout)
    const int mbase = (lane >> 4) * 8;     // C/D: VGPR v -> M = v (+8 upper half)
    const int b0    = blockIdx.x * 16;     // batch row base
    const int c0    = wave * 16;           // this wave's h-column tile base

    // zero initial hidden state
    for (int i = tid; i < 16 * HSTR; i += 512) hbuf[0][i] = 0.0f;

    // loop-invariant per-lane biases
    const float bias_r = b_hh[      c0 + ln];
    const float bias_z = b_hh[256 + c0 + ln];
    const float bias_n = b_hh[512 + c0 + ln];
    float bhead = 0.0f;
    if (wave < 2) { int n = wave * 16 + ln; bhead = (n < NO_) ? b_head[n] : 0.0f; }

    // loop-invariant B-operand row pointers (W_hhT rows are contiguous in K)
    const float* __restrict__ wr = W_hhT + (size_t)(      c0 + ln) * H_;
    const float* __restrict__ wz = W_hhT + (size_t)(256 + c0 + ln) * H_;
    const float* __restrict__ wn = W_hhT + (size_t)(512 + c0 + ln) * H_;

    int cur = 0;
    for (int t = 0; t < T_; ++t) {
        // ---- phase 1: fetch this step's 16 tokens ----
        if (tid < 16) tokbuf[tid] = tokens[(b0 + tid) * T_ + t];
        __syncthreads();

        // ---- phase 2: gh = h @ W_hh for the 3 gate tiles of this wave ----
        v8f ar = {}; v8f az = {}; v8f an = {};
        {
            const float* ha = &hbuf[cur][ln * HSTR];      // A: M = ln, K pair by lane half
            #pragma unroll 4
            for (int k0 = 0; k0 < H_; k0 += 4) {
                v2f a  = *(const v2f*)&ha[k0 + khalf];    // ds_load_b64, conflict-free
                v2f br = *(const v2f*)&wr[k0 + khalf];    // global b64 (L2-resident W)
                v2f bz = *(const v2f*)&wz[k0 + khalf];
                v2f bn = *(const v2f*)&wn[k0 + khalf];
                ar = __builtin_amdgcn_wmma_f32_16x16x4_f32(false, a, false, br, (short)0, ar, false, false);
                az = __builtin_amdgcn_wmma_f32_16x16x4_f32(false, a, false, bz, (short)0, az, false, false);
                an = __builtin_amdgcn_wmma_f32_16x16x4_f32(false, a, false, bn, (short)0, an, false, false);
            }
        }

        // ---- gates + h update (elementwise on C/D layout) ----
        #pragma unroll
        for (int v = 0; v < 8; ++v) {
            const int m = mbase + v;
            const float* gi_m = gi_table + (size_t)tokbuf[m] * G3;   // gather via gi table
            const float i_r = gi_m[      c0 + ln];
            const float i_z = gi_m[256 + c0 + ln];
            const float i_n = gi_m[512 + c0 + ln];
            const float r  = sigf(i_r + ar[v] + bias_r);
            const float z  = sigf(i_z + az[v] + bias_z);
            const float nn = tanh_f(i_n + r * (an[v] + bias_n));
            const float ho = hbuf[cur][m * HSTR + c0 + ln];
            const float hn = (1.0f - z) * nn + z * ho;
            hbuf[cur ^ 1][m * HSTR + c0 + ln] = hn;
            out_betas[((size_t)(b0 + m) * T_ + t) * H_ + c0 + ln] = z;  // beta = z
        }
        __syncthreads();
        cur ^= 1;

        // ---- phase 3: logits = h_new @ W_head (+bias), waves 0 and 1 ----
        if (wave < 2) {
            const int n0 = wave * 16;
            v8f acc = {};
            const float* wh = W_headT + (size_t)(n0 + ln) * H_;
            const float* ha = &hbuf[cur][ln * HSTR];
            #pragma unroll 4
            for (int k0 = 0; k0 < H_; k0 += 4) {
                v2f a = *(const v2f*)&ha[k0 + khalf];
                v2f b = *(const v2f*)&wh[k0 + khalf];
                acc = __builtin_amdgcn_wmma_f32_16x16x4_f32(false, a, false, b, (short)0, acc, false, false);
            }
            const int n = n0 + ln;
            if (n < NO_) {
                #pragma unroll
                for (int v = 0; v < 8; ++v) {
                    const int m = mbase + v;
                    out_logits[((size_t)(b0 + m) * T_ + t) * NO_ + n] = acc[v] + bhead;
                }
            }
        }
        // no barrier needed here: next phase1 only touches tokbuf (read after next barrier)
    }
}

// ---------------------------------------------------------------------------
extern "C" void kernel_launch(void* const* d_in, const int* in_sizes, int n_in,
                              void* d_out, int out_size, void* d_ws, size_t ws_size,
                              hipStream_t stream)
{
    (void)in_sizes; (void)n_in; (void)out_size; (void)ws_size;
    const int*   tokens = (const int*)  d_in[0];
    const float* embed  = (const float*)d_in[1];
    const float* W_ih   = (const float*)d_in[2];
    const float* W_hh   = (const float*)d_in[3];
    const float* b_ih   = (const float*)d_in[4];
    const float* b_hh   = (const float*)d_in[5];
    const float* W_head = (const float*)d_in[6];
    const float* b_head = (const float*)d_in[7];

    float* ws       = (float*)d_ws;
    float* gi_table = ws;                          // 18*768   floats
    float* W_hhT    = gi_table + NV_ * G3;         // 768*256  floats
    float* W_headT  = W_hhT + (size_t)G3 * H_;     // 32*256   floats  (~0.9 MB total)

    float* out_logits = (float*)d_out;
    float* out_betas  = out_logits + (size_t)B_ * T_ * NO_;

    prep_gi_table<<<NV_, 256, 0, stream>>>(embed, W_ih, b_ih, gi_table);
    prep_whhT   <<<(G3 * H_ + 255) / 256, 256, 0, stream>>>(W_hh, W_hhT);
    prep_wheadT <<<(NOP_ * H_ + 255) / 256, 256, 0, stream>>>(W_head, W_headT);
    mogru_scan  <<<2, 512, 0, stream>>>(tokens, gi_table, W_hhT, b_hh,
                                        W_headT, b_head, out_logits, out_betas);
}